// NT_Xent_33681133535421
// MI455X (gfx1250) — compile-verified
//
#include <hip/hip_runtime.h>
#include <hip/hip_bf16.h>

// NT-Xent fused: f32 WMMA GEMM (16x16x4) + online softmax stats, never
// materializing the 4096x4096 similarity matrix. Each wave computes two
// 16x16 column tiles per K-pass (A-fragment reuse), branchless epilogue.

#define NN 4096
#define BB 2048
#define DD 512

typedef float v2f __attribute__((ext_vector_type(2)));
typedef float v8f __attribute__((ext_vector_type(8)));

__device__ __forceinline__ const float* rowptr(const float* __restrict__ zi,
                                               const float* __restrict__ zj,
                                               int row) {
  return (row < BB) ? (zi + row * DD) : (zj + (row - BB) * DD);
}

// ---------------- kernel 1: inverse row norms ----------------
__global__ __launch_bounds__(32) void ntxent_norm(const float* __restrict__ zi,
                                                  const float* __restrict__ zj,
                                                  float* __restrict__ inv) {
  const int row = blockIdx.x;
  const int lane = threadIdx.x;
  const float* zr = rowptr(zi, zj, row);
  float ss = 0.0f;
  for (int i = lane * 4; i < DD; i += 128) {
    float4 v = *(const float4*)(zr + i);
    ss += v.x * v.x + v.y * v.y + v.z * v.z + v.w * v.w;
  }
  for (int off = 16; off > 0; off >>= 1) ss += __shfl_xor(ss, off, 32);
  if (lane == 0) inv[row] = 1.0f / fmaxf(sqrtf(ss), 1e-8f);
}

// ---------------- kernel 2: positive-pair logits ----------------
__global__ __launch_bounds__(32) void ntxent_pos(const float* __restrict__ zi,
                                                 const float* __restrict__ zj,
                                                 const float* __restrict__ tptr,
                                                 const float* __restrict__ inv,
                                                 float* __restrict__ pos) {
  const int row = blockIdx.x;
  const int lane = threadIdx.x;
  const int pm = (row + BB) & (NN - 1);
  const float* a = rowptr(zi, zj, row);
  const float* b = rowptr(zi, zj, pm);
  float d = 0.0f;
  for (int i = lane * 4; i < DD; i += 128) {
    float4 x = *(const float4*)(a + i);
    float4 y = *(const float4*)(b + i);
    d += x.x * y.x + x.y * y.y + x.z * y.z + x.w * y.w;
  }
  for (int off = 16; off > 0; off >>= 1) d += __shfl_xor(d, off, 32);
  if (lane == 0) pos[row] = d * inv[row] * inv[pm] / tptr[0];
}

// ---------------- kernel 3: fused WMMA GEMM + online softmax stats ----------------
// block = 128 threads (4 wave32s); block owns 16 rows; each wave streams pairs of
// 16-wide column tiles (two accumulators share one A fragment per WMMA step).
__global__ __launch_bounds__(128) void ntxent_main(const float* __restrict__ zi,
                                                   const float* __restrict__ zj,
                                                   const float* __restrict__ tptr,
                                                   const float* __restrict__ inv,
                                                   const float* __restrict__ pos,
                                                   float* __restrict__ lse,
                                                   int* __restrict__ cnt) {
  __shared__ float Alds[16][514];  // pad 2 floats: conflict-free + 8B aligned b64 reads
  __shared__ float sM[4][16];
  __shared__ float sS[4][16];
  __shared__ int   sC[4][16];

  const int rb   = blockIdx.x;     // rows rb*16 .. rb*16+15
  const int tid  = threadIdx.x;
  const int wave = tid >> 5;
  const int lane = tid & 31;
  const int l16  = lane & 15;
  const int half = lane >> 4;
  const int row0 = rb * 16;
  const float invT = 1.0f / tptr[0];

  // Cooperative A-tile load: 16 rows x 512 (contiguous: row0 % 16 == 0, halves are 2048)
  {
    const float* src = rowptr(zi, zj, row0);
    for (int i = tid; i < 16 * 128; i += 128) {
      const int r  = i >> 7;       // row 0..15
      const int c4 = i & 127;      // float4 index within row
      const float4 v = ((const float4*)(src + r * DD))[c4];
      float* dst = &Alds[r][c4 * 4];
      dst[0] = v.x; dst[1] = v.y; dst[2] = v.z; dst[3] = v.w;
    }
  }
  __syncthreads();

  // Per-lane row constants: C/D layout row = v + 8*half, col = cb + l16
  float rinv[8], pv[8];
  int grv[8], pmv[8];
#pragma unroll
  for (int v = 0; v < 8; ++v) {
    const int gr = row0 + v + 8 * half;
    grv[v] = gr;
    pmv[v] = (gr + BB) & (NN - 1);
    rinv[v] = inv[gr];
    pv[v] = pos[gr];
  }

  float m[8], s[8];
  int c[8];
#pragma unroll
  for (int v = 0; v < 8; ++v) { m[v] = -1e30f; s[v] = 0.0f; c[v] = 0; }

  const int koff = half * 2;  // A/B frag: low lanes K={0,1}, high lanes K={2,3}

  // Branchless online-softmax tile epilogue (keeps EXEC full; cndmask only)
  auto epilogue = [&](const v8f& acc, int cb) {
    const int gcol = cb + l16;
    const float cinv = inv[gcol];
#pragma unroll
    for (int v = 0; v < 8; ++v) {
      const float sv = acc[v] * rinv[v] * cinv * invT;
      const bool notdiag = (gcol != grv[v]);
      const float nm  = notdiag ? fmaxf(m[v], sv) : m[v];
      const float add = notdiag ? __expf(sv - nm) : 0.0f;
      s[v] = s[v] * __expf(m[v] - nm) + add;
      m[v] = nm;
      c[v] += (notdiag && (sv > pv[v]) && (gcol != pmv[v])) ? 1 : 0;
    }
  };

  // 128 pairs of adjacent 16-wide column tiles, round-robined over 4 waves
  for (int p = wave; p < NN / 32; p += 4) {
    const int cb0 = p * 32;
    const float* brow0 = rowptr(zi, zj, cb0 + l16);        // B col N = l16
    const float* brow1 = rowptr(zi, zj, cb0 + 16 + l16);
    v8f acc0 = {0.0f, 0.0f, 0.0f, 0.0f, 0.0f, 0.0f, 0.0f, 0.0f};
    v8f acc1 = {0.0f, 0.0f, 0.0f, 0.0f, 0.0f, 0.0f, 0.0f, 0.0f};
#pragma unroll 4
    for (int k0 = 0; k0 < DD; k0 += 4) {
      v2f a  = *(const v2f*)&Alds[l16][k0 + koff];          // A row M = l16 (shared)
      v2f b0 = *(const v2f*)(brow0 + k0 + koff);
      v2f b1 = *(const v2f*)(brow1 + k0 + koff);
      acc0 = __builtin_amdgcn_wmma_f32_16x16x4_f32(
          false, a, false, b0, (short)0, acc0, false, false);
      acc1 = __builtin_amdgcn_wmma_f32_16x16x4_f32(
          false, a, false, b1, (short)0, acc1, false, false);
    }
    epilogue(acc0, cb0);
    epilogue(acc1, cb0 + 16);
  }

  // Reduce across the 16 lanes of each half (same row, different columns)
#pragma unroll
  for (int v = 0; v < 8; ++v) {
    for (int off = 1; off < 16; off <<= 1) {
      const float om = __shfl_xor(m[v], off, 16);
      const float os = __shfl_xor(s[v], off, 16);
      const int   oc = __shfl_xor(c[v], off, 16);
      const float nm = fmaxf(m[v], om);
      s[v] = s[v] * __expf(m[v] - nm) + os * __expf(om - nm);
      m[v] = nm;
      c[v] += oc;
    }
    if (l16 == 0) {
      const int r = v + 8 * half;
      sM[wave][r] = m[v];
      sS[wave][r] = s[v];
      sC[wave][r] = c[v];
    }
  }
  __syncthreads();

  // Merge the 4 waves' column partitions and finalize per-row results
  if (tid < 16) {
    float M = sM[0][tid], S = sS[0][tid];
    int C = sC[0][tid];
#pragma unroll
    for (int w = 1; w < 4; ++w) {
      const float om = sM[w][tid], os = sS[w][tid];
      const float nm = fmaxf(M, om);
      S = S * __expf(M - nm) + os * __expf(om - nm);
      M = nm;
      C += sC[w][tid];
    }
    lse[row0 + tid] = __logf(S) + M;
    cnt[row0 + tid] = C;
  }
}

// ---------------- kernel 4: final reduction ----------------
__global__ __launch_bounds__(256) void ntxent_final(const float* __restrict__ lse,
                                                    const float* __restrict__ pos,
                                                    const int* __restrict__ cnt,
                                                    float* __restrict__ out) {
  __shared__ float sl[256];
  __shared__ float sc[256];
  const int tid = threadIdx.x;
  float l = 0.0f, cc = 0.0f;
  for (int n = tid; n < NN; n += 256) {
    l += lse[n] - pos[n];
    cc += (float)cnt[n];
  }
  sl[tid] = l; sc[tid] = cc;
  __syncthreads();
  for (int st = 128; st > 0; st >>= 1) {
    if (tid < st) { sl[tid] += sl[tid + st]; sc[tid] += sc[tid + st]; }
    __syncthreads();
  }
  if (tid == 0) {
    out[0] = sl[0] / (float)NN;   // loss
    out[1] = sc[0] / (float)NN;   // avg_rank
  }
}

extern "C" void kernel_launch(void* const* d_in, const int* in_sizes, int n_in,
                              void* d_out, int out_size, void* d_ws, size_t ws_size,
                              hipStream_t stream) {
  const float* zi   = (const float*)d_in[0];  // [2048,512] f32
  const float* zj   = (const float*)d_in[1];  // [2048,512] f32
  const float* tptr = (const float*)d_in[2];  // [1] f32 (temperature)
  float* out = (float*)d_out;                 // [loss, avg_rank]

  float* wsf = (float*)d_ws;
  float* inv = wsf;                   // [4096]
  float* pos = wsf + NN;              // [4096]
  float* lse = wsf + 2 * NN;          // [4096]
  int*   cnt = (int*)(wsf + 3 * NN);  // [4096]

  ntxent_norm<<<NN, 32, 0, stream>>>(zi, zj, inv);
  ntxent_pos<<<NN, 32, 0, stream>>>(zi, zj, tptr, inv, pos);
  ntxent_main<<<NN / 16, 128, 0, stream>>>(zi, zj, tptr, inv, pos, lse, cnt);
  ntxent_final<<<1, 256, 0, stream>>>(lse, pos, cnt, out);
}